// InternLMDecoderLayer_68676527063337
// MI455X (gfx1250) — compile-verified
//
#include <hip/hip_runtime.h>
#include <hip/hip_bf16.h>
#include <math.h>

#define B_      16
#define S_      1024
#define H_      32
#define D_      128
#define HIDDEN  4096
#define INTER   11008
#define QKV_N   12288
#define GU_N    22016
#define EPS_    1e-6f
#define SCALING 0.08838834764831845f  /* 128^-0.5 */

typedef float v2f __attribute__((ext_vector_type(2)));
typedef float v8f __attribute__((ext_vector_type(8)));

// ---------------------------------------------------------------------------
// RMSNorm: one block per batch row
// ---------------------------------------------------------------------------
__global__ void rmsnorm_kernel(const float* __restrict__ x,
                               const float* __restrict__ w,
                               float* __restrict__ out) {
  const int b = blockIdx.x;
  const float* row = x + b * HIDDEN;
  __shared__ float red[256];
  float s = 0.f;
  for (int i = threadIdx.x; i < HIDDEN; i += 256) {
    float v = row[i];
    s += v * v;
  }
  red[threadIdx.x] = s;
  __syncthreads();
  for (int off = 128; off > 0; off >>= 1) {
    if (threadIdx.x < off) red[threadIdx.x] += red[threadIdx.x + off];
    __syncthreads();
  }
  const float inv = rsqrtf(red[0] / (float)HIDDEN + EPS_);
  for (int i = threadIdx.x; i < HIDDEN; i += 256)
    out[b * HIDDEN + i] = row[i] * inv * w[i];
}

// ---------------------------------------------------------------------------
// 16xN fp32 GEMM via V_WMMA_F32_16X16X4_F32.
// One wave = one 16x32 output tile (2 accumulators, shared A fetch).
// 4 waves / block.  A ([16,K], 688KB max) is L2-resident.
// W: [K,N] row-major, streamed once with prefetch.  out: [16,N].
// Optional bias[N] and residual [16,N].
// ---------------------------------------------------------------------------
__global__ void gemm16_wmma(const float* __restrict__ A,
                            const float* __restrict__ W,
                            const float* __restrict__ bias,
                            const float* __restrict__ resid,
                            float* __restrict__ out, int K, int N) {
  const int lane = threadIdx.x & 31;
  const int wave = threadIdx.x >> 5;
  const int n0   = (blockIdx.x * 4 + wave) * 32;
  const int kb   = (lane >> 4) * 2;      // lanes 0-15 -> K0/K1, 16-31 -> K2/K3
  const int nn   = n0 + (lane & 15);
  const float* arow = A + (lane & 15) * K + kb;
  const float* w0 = W + nn;
  const float* w1 = W + nn + 16;
  v8f c0 = {};
  v8f c1 = {};
  for (int k = 0; k < K; k += 16) {
    // deep prefetch of the weight stream (speculative: OOB silently dropped)
    __builtin_prefetch(w0 + (size_t)(k + kb + 256) * N, 0, 1);
#pragma unroll
    for (int kk = 0; kk < 16; kk += 4) {
      const int kc = k + kk;
      v2f av;
      av.x = arow[kc];
      av.y = arow[kc + 1];
      const size_t r0 = (size_t)(kc + kb) * N;
      const size_t r1 = (size_t)(kc + kb + 1) * N;
      v2f b0;
      b0.x = w0[r0];
      b0.y = w0[r1];
      v2f b1;
      b1.x = w1[r0];
      b1.y = w1[r1];
      c0 = __builtin_amdgcn_wmma_f32_16x16x4_f32(false, av, false, b0,
                                                 (short)0, c0, false, false);
      c1 = __builtin_amdgcn_wmma_f32_16x16x4_f32(false, av, false, b1,
                                                 (short)0, c1, false, false);
    }
  }
  const float badd0 = bias ? bias[nn] : 0.f;
  const float badd1 = bias ? bias[nn + 16] : 0.f;
  const int rbase = (lane >> 4) * 8;     // C/D layout: VGPR v -> row v (+8 hi lanes)
#pragma unroll
  for (int v = 0; v < 8; ++v) {
    const int row = rbase + v;
    float v0 = c0[v] + badd0;
    float v1 = c1[v] + badd1;
    if (resid) {
      v0 += resid[row * N + nn];
      v1 += resid[row * N + nn + 16];
    }
    out[row * N + nn]      = v0;
    out[row * N + nn + 16] = v1;
  }
}

// ---------------------------------------------------------------------------
// Fused gate/up projection + SiLU(gate)*up.  16x32 tile per wave,
// 4 WMMA accumulators sharing one A fetch.
// A: [16,HIDDEN], W: [HIDDEN, 2*INTER], act: [16, INTER]
// ---------------------------------------------------------------------------
__global__ void gateup_wmma(const float* __restrict__ A,
                            const float* __restrict__ W,
                            float* __restrict__ act) {
  const int lane = threadIdx.x & 31;
  const int wave = threadIdx.x >> 5;
  const int n0   = (blockIdx.x * 4 + wave) * 32;
  const int kb   = (lane >> 4) * 2;
  const int ng   = n0 + (lane & 15);
  const float* arow = A + (lane & 15) * HIDDEN + kb;
  const float* wg0 = W + ng;
  const float* wg1 = W + ng + 16;
  const float* wu0 = W + ng + INTER;
  const float* wu1 = W + ng + INTER + 16;
  v8f cg0 = {}, cg1 = {}, cu0 = {}, cu1 = {};
  for (int k = 0; k < HIDDEN; k += 16) {
    __builtin_prefetch(wg0 + (size_t)(k + kb + 128) * GU_N, 0, 1);
    __builtin_prefetch(wu0 + (size_t)(k + kb + 128) * GU_N, 0, 1);
#pragma unroll
    for (int kk = 0; kk < 16; kk += 4) {
      const int kc = k + kk;
      v2f av;
      av.x = arow[kc];
      av.y = arow[kc + 1];
      const size_t r0 = (size_t)(kc + kb) * GU_N;
      const size_t r1 = (size_t)(kc + kb + 1) * GU_N;
      v2f bg0; bg0.x = wg0[r0]; bg0.y = wg0[r1];
      v2f bg1; bg1.x = wg1[r0]; bg1.y = wg1[r1];
      v2f bu0; bu0.x = wu0[r0]; bu0.y = wu0[r1];
      v2f bu1; bu1.x = wu1[r0]; bu1.y = wu1[r1];
      cg0 = __builtin_amdgcn_wmma_f32_16x16x4_f32(false, av, false, bg0,
                                                  (short)0, cg0, false, false);
      cg1 = __builtin_amdgcn_wmma_f32_16x16x4_f32(false, av, false, bg1,
                                                  (short)0, cg1, false, false);
      cu0 = __builtin_amdgcn_wmma_f32_16x16x4_f32(false, av, false, bu0,
                                                  (short)0, cu0, false, false);
      cu1 = __builtin_amdgcn_wmma_f32_16x16x4_f32(false, av, false, bu1,
                                                  (short)0, cu1, false, false);
    }
  }
  const int rbase = (lane >> 4) * 8;
#pragma unroll
  for (int v = 0; v < 8; ++v) {
    const int row = rbase + v;
    const float g0 = cg0[v], g1 = cg1[v];
    const float u0 = cu0[v], u1 = cu1[v];
    const float s0 = g0 / (1.f + __expf(-g0));
    const float s1 = g1 / (1.f + __expf(-g1));
    act[row * INTER + ng]      = s0 * u0;
    act[row * INTER + ng + 16] = s1 * u1;
  }
}

// ---------------------------------------------------------------------------
// RoPE (GPT-NeoX) applied in place to q and k inside the qkv buffer.
// One block per (b,h), 64 threads, each handling the pair (i, i+64).
// ---------------------------------------------------------------------------
__global__ void rope_kernel(float* __restrict__ qkv,
                            const int* __restrict__ positions) {
  const int bh = blockIdx.x;
  const int b = bh >> 5;
  const int h = bh & 31;
  const int i = threadIdx.x;                          // 0..63
  const float posf = (float)positions[b];
  const float freq = __powf(10000.f, -(2.f * (float)i) / (float)D_);
  float sv, cv;
  __sincosf(posf * freq, &sv, &cv);
  float* q = qkv + b * QKV_N + h * D_;
  float* k = q + HIDDEN;
  const float q1 = q[i], q2 = q[i + 64];
  q[i]      = q1 * cv - q2 * sv;
  q[i + 64] = q2 * cv + q1 * sv;
  const float k1 = k[i], k2 = k[i + 64];
  k[i]      = k1 * cv - k2 * sv;
  k[i + 64] = k2 * cv + k1 * sv;
}

// ---------------------------------------------------------------------------
// Decode attention. One block per (b,h), 8 waves. The cache is NOT written:
// at s == pos we substitute the fresh RoPE'd k/v from the qkv buffer
// (equivalent to reference's cache-write-then-read).
// ---------------------------------------------------------------------------
__global__ void attn_kernel(const float* __restrict__ qkv,
                            const float* __restrict__ k_cache,
                            const float* __restrict__ v_cache,
                            const int* __restrict__ positions,
                            float* __restrict__ out) {
  const int b = blockIdx.x;
  const int h = blockIdx.y;
  const int pos = positions[b];
  const int L = pos + 1;
  const int lane = threadIdx.x & 31;
  const int wave = threadIdx.x >> 5;                  // 0..7

  __shared__ float sc[S_];
  __shared__ float acc[8][D_];
  __shared__ float wmax[8];
  __shared__ float wsum[8];

  const float* qp = qkv + b * QKV_N + h * D_;
  const float* kn = qp + HIDDEN;                      // fresh k for s == pos
  const float* vn = qp + 2 * HIDDEN;                  // fresh v for s == pos
  const float4 q4 = *(const float4*)(qp + lane * 4);

  // --- scores: one wave per key, lane-split dot over D=128 ---
  for (int s = wave; s < L; s += 8) {
    const float* kp = (s == pos)
        ? kn
        : (k_cache + ((size_t)(b * S_ + s)) * (H_ * D_) + h * D_);
    const float4 k4 = *(const float4*)(kp + lane * 4);
    float d = q4.x * k4.x + q4.y * k4.y + q4.z * k4.z + q4.w * k4.w;
    for (int off = 16; off > 0; off >>= 1) d += __shfl_xor(d, off, 32);
    if (lane == 0) sc[s] = d * SCALING;
  }
  __syncthreads();

  // --- block max ---
  float m = -1e30f;
  for (int i = threadIdx.x; i < L; i += 256) m = fmaxf(m, sc[i]);
  for (int off = 16; off > 0; off >>= 1) m = fmaxf(m, __shfl_xor(m, off, 32));
  if (lane == 0) wmax[wave] = m;
  __syncthreads();
  m = wmax[0];
#pragma unroll
  for (int w = 1; w < 8; ++w) m = fmaxf(m, wmax[w]);

  // --- exp + block sum (same i-mapping as max loop: no sc hazards) ---
  float ssum = 0.f;
  for (int i = threadIdx.x; i < L; i += 256) {
    const float e = __expf(sc[i] - m);
    sc[i] = e;
    ssum += e;
  }
  for (int off = 16; off > 0; off >>= 1) ssum += __shfl_xor(ssum, off, 32);
  if (lane == 0) wsum[wave] = ssum;
  __syncthreads();
  ssum = 0.f;
#pragma unroll
  for (int w = 0; w < 8; ++w) ssum += wsum[w];
  const float inv = 1.f / ssum;

  // --- weighted V accumulation: wave-split over s, lane-split over d ---
  float4 a4 = make_float4(0.f, 0.f, 0.f, 0.f);
  for (int s = wave; s < L; s += 8) {
    const float* vp = (s == pos)
        ? vn
        : (v_cache + ((size_t)(b * S_ + s)) * (H_ * D_) + h * D_);
    const float4 v4 = *(const float4*)(vp + lane * 4);
    const float wgt = sc[s];
    a4.x += wgt * v4.x;
    a4.y += wgt * v4.y;
    a4.z += wgt * v4.z;
    a4.w += wgt * v4.w;
  }
  acc[wave][lane * 4 + 0] = a4.x;
  acc[wave][lane * 4 + 1] = a4.y;
  acc[wave][lane * 4 + 2] = a4.z;
  acc[wave][lane * 4 + 3] = a4.w;
  __syncthreads();

  if (threadIdx.x < D_) {
    float t = 0.f;
#pragma unroll
    for (int w = 0; w < 8; ++w) t += acc[w][threadIdx.x];
    out[b * HIDDEN + h * D_ + threadIdx.x] = t * inv;
  }
}

// ---------------------------------------------------------------------------
extern "C" void kernel_launch(void* const* d_in, const int* in_sizes, int n_in,
                              void* d_out, int out_size, void* d_ws, size_t ws_size,
                              hipStream_t stream) {
  (void)in_sizes; (void)n_in; (void)out_size; (void)ws_size;

  const int*   positions     = (const int*)d_in[0];
  const float* hidden_states = (const float*)d_in[1];
  const float* k_cache       = (const float*)d_in[2];
  const float* v_cache       = (const float*)d_in[3];
  const float* qkv_w         = (const float*)d_in[4];
  const float* qkv_b         = (const float*)d_in[5];
  const float* o_w           = (const float*)d_in[6];
  const float* o_b           = (const float*)d_in[7];
  const float* gate_up_w     = (const float*)d_in[8];
  const float* down_w        = (const float*)d_in[9];
  const float* ln1_w         = (const float*)d_in[10];
  const float* ln2_w         = (const float*)d_in[11];

  float* ws       = (float*)d_ws;
  float* xn1      = ws;                         // [16, 4096]
  float* qkv      = xn1 + B_ * HIDDEN;          // [16, 12288]
  float* attn_out = qkv + B_ * QKV_N;           // [16, 4096]
  float* hidden   = attn_out + B_ * HIDDEN;     // [16, 4096]
  float* xn2      = hidden + B_ * HIDDEN;       // [16, 4096]
  float* act      = xn2 + B_ * HIDDEN;          // [16, 11008]
  float* outp     = (float*)d_out;              // [16, 4096]

  rmsnorm_kernel<<<B_, 256, 0, stream>>>(hidden_states, ln1_w, xn1);

  // QKV: [16,4096] x [4096,12288] (+bias)   384 waves / 4 = 96 blocks
  gemm16_wmma<<<QKV_N / 128, 128, 0, stream>>>(xn1, qkv_w, qkv_b, nullptr,
                                               qkv, HIDDEN, QKV_N);

  rope_kernel<<<B_ * H_, 64, 0, stream>>>(qkv, positions);

  attn_kernel<<<dim3(B_, H_), 256, 0, stream>>>(qkv, k_cache, v_cache,
                                                positions, attn_out);

  // O proj: [16,4096] x [4096,4096] (+bias, +residual hidden_states)
  gemm16_wmma<<<HIDDEN / 128, 128, 0, stream>>>(attn_out, o_w, o_b,
                                                hidden_states, hidden,
                                                HIDDEN, HIDDEN);

  rmsnorm_kernel<<<B_, 256, 0, stream>>>(hidden, ln2_w, xn2);

  // Fused gate/up + SiLU: 344 waves / 4 = 86 blocks
  gateup_wmma<<<INTER / 128, 128, 0, stream>>>(xn2, gate_up_w, act);

  // Down proj: [16,11008] x [11008,4096] (+residual hidden) -> output
  gemm16_wmma<<<HIDDEN / 128, 128, 0, stream>>>(act, down_w, nullptr,
                                                hidden, outp, INTER, HIDDEN);
}